// fusion_9990093931238
// MI455X (gfx1250) — compile-verified
//
#include <hip/hip_runtime.h>

typedef __attribute__((ext_vector_type(2))) float v2f;
typedef __attribute__((ext_vector_type(4))) float v4f;
typedef __attribute__((ext_vector_type(8))) float v8f;

constexpr int B = 8, N = 4096, C = 1024;
constexpr int TOK   = 64;        // tokens per reduce block
constexpr int SLABS = N / TOK;   // 64
constexpr int WAVES = 8;         // 256 threads / wave32

__global__ void zero_kernel(float* ws, int n) {
    int i = blockIdx.x * blockDim.x + threadIdx.x;
    if (i < n) ws[i] = 0.0f;
}

// Fused: per-token channel sums (phase 1) + WMMA GEMV partials (phase 2).
// Phase 2 re-reads the same 0.5 MB slab -> L2/L1 hits, so HBM sees data once.
__global__ __launch_bounds__(256) void reduce_kernel(const float* __restrict__ f1,
                                                     const float* __restrict__ f2,
                                                     float* __restrict__ Crow,
                                                     float* __restrict__ Ccol) {
    __shared__ float s1[TOK];
    __shared__ float s2[TOK];
    const int batch = blockIdx.x / SLABS;
    const int n0    = (blockIdx.x % SLABS) * TOK;
    const int lane  = threadIdx.x & 31;
    const int wave  = threadIdx.x >> 5;
    const float* f1b = f1 + (size_t)batch * N * C;
    const float* f2b = f2 + (size_t)batch * N * C;

    // ---- Phase 1: one wave per token row, 8 tokens per wave ----
    for (int j = 0; j < TOK / WAVES; ++j) {
        const int tt = wave * (TOK / WAVES) + j;
        const v4f* r1 = (const v4f*)(f1b + (size_t)(n0 + tt) * C);
        const v4f* r2 = (const v4f*)(f2b + (size_t)(n0 + tt) * C);
        float a = 0.0f, b = 0.0f;
        for (int k = lane; k < C / 4; k += 32) {
            v4f x = r1[k]; a += (x.x + x.y) + (x.z + x.w);
            v4f y = r2[k]; b += (y.x + y.y) + (y.z + y.w);
        }
        for (int m = 16; m >= 1; m >>= 1) {
            a += __shfl_xor(a, m, 32);
            b += __shfl_xor(b, m, 32);
        }
        if (lane == 0) { s1[tt] = a; s2[tt] = b; }
    }
    __syncthreads();

    // ---- Phase 2: WMMA f32 16x16x4 GEMV partials ----
    // A[i,k] = s (replicated over rows i); B[k,j] = f[n0+k, c0+j].
    // A 16x4 layout: lanes 0-15 hold K=0 (v0), K=1 (v1); lanes 16-31 K=2,3.
    // Same packing rule used for B (4x16), so the k-pairing is consistent.
    const int klo = (lane >> 4) << 1;   // 0 or 2
    const int cl  = lane & 15;
    for (int t = 0; t < 8; ++t) {
        const int c0 = wave * 128 + t * 16;
        v8f acc1 = {};   // partial C_row (f1 * s2)
        v8f acc2 = {};   // partial C_col (f2 * s1)
        for (int s = 0; s < TOK / 4; ++s) {
            const int kb = s * 4 + klo;
            v2f aRow, aCol, bm1, bm2;
            aRow.x = s2[kb]; aRow.y = s2[kb + 1];
            aCol.x = s1[kb]; aCol.y = s1[kb + 1];
            const float* p1 = f1b + (size_t)(n0 + kb) * C + c0 + cl;
            const float* p2 = f2b + (size_t)(n0 + kb) * C + c0 + cl;
            bm1.x = p1[0]; bm1.y = p1[C];
            bm2.x = p2[0]; bm2.y = p2[C];
            acc1 = __builtin_amdgcn_wmma_f32_16x16x4_f32(
                false, aRow, false, bm1, (short)0, acc1, false, false);
            acc2 = __builtin_amdgcn_wmma_f32_16x16x4_f32(
                false, aCol, false, bm2, (short)0, acc2, false, false);
        }
        // All 16 D rows are identical; row 0 lives in v0 of lanes 0-15.
        if (lane < 16) {
            atomicAdd(&Crow[batch * C + c0 + cl], acc1[0]);
            atomicAdd(&Ccol[batch * C + c0 + cl], acc2[0]);
        }
    }
}

// In-place: x -> 1 - sigmoid(x) = 1/(1+exp(x)).  Re-zeroed every call, so
// overwriting the accumulators is deterministic across graph replays.
__global__ void scale_kernel(float* Crow, float* Ccol) {
    int i = blockIdx.x * blockDim.x + threadIdx.x;
    if (i < B * C) {
        Crow[i] = 1.0f / (1.0f + __expf(Crow[i]));
        Ccol[i] = 1.0f / (1.0f + __expf(Ccol[i]));
    }
}

// Streaming pass: regular loads (may hit residual L2 lines from reduce pass),
// non-temporal stores (outputs are never re-read; keep them from thrashing
// the 192 MB L2 -> global_store_b128 with TH=NT).
__global__ __launch_bounds__(256) void apply_kernel(const v4f* __restrict__ f1,
                                                    const v4f* __restrict__ f2,
                                                    const v4f* __restrict__ sc1,
                                                    const v4f* __restrict__ sc2,
                                                    v4f* __restrict__ o1,
                                                    v4f* __restrict__ o2) {
    size_t i = (size_t)blockIdx.x * blockDim.x + threadIdx.x;   // v4f index
    const int c4    = (int)(i & (C / 4 - 1));                   // C/4 = 256
    const int batch = (int)(i >> 20);                           // N*C/4 = 2^20
    const v4f sa = sc1[batch * (C / 4) + c4];                   // L2-resident 32KB
    const v4f sb = sc2[batch * (C / 4) + c4];
    v4f a = f1[i];
    v4f b = f2[i];
    a *= sa;
    b *= sb;
    __builtin_nontemporal_store(a, &o1[i]);
    __builtin_nontemporal_store(b, &o2[i]);
}

extern "C" void kernel_launch(void* const* d_in, const int* in_sizes, int n_in,
                              void* d_out, int out_size, void* d_ws, size_t ws_size,
                              hipStream_t stream) {
    const float* f1 = (const float*)d_in[0];
    const float* f2 = (const float*)d_in[1];
    float* out  = (float*)d_out;
    float* Crow = (float*)d_ws;            // [B,C]
    float* Ccol = Crow + B * C;            // [B,C]  (64 KB total scratch)

    zero_kernel<<<(2 * B * C + 255) / 256, 256, 0, stream>>>(Crow, 2 * B * C);
    reduce_kernel<<<B * SLABS, 256, 0, stream>>>(f1, f2, Crow, Ccol);
    scale_kernel<<<(B * C + 255) / 256, 256, 0, stream>>>(Crow, Ccol);

    const size_t q = (size_t)B * N * C / 4;   // v4f elements per tensor
    apply_kernel<<<(unsigned)(q / 256), 256, 0, stream>>>(
        (const v4f*)f1, (const v4f*)f2,
        (const v4f*)Crow, (const v4f*)Ccol,
        (v4f*)out, (v4f*)out + q);
}